// IntraSectorGAT_23733989277957
// MI455X (gfx1250) — compile-verified
//
#include <hip/hip_runtime.h>
#include <hip/hip_bf16.h>

// ---------------------------------------------------------------------------
// GAT layer for MI455X (gfx1250, wave32).
//   h = x @ W            -> V_WMMA_F32_16X16X4_F32, 4 M-tiles per wave,
//                           B fragments as b64 loads from transposed W
//   a_src/a_dst          -> wave shuffle reductions
//   softmax over edges   -> ordered-uint atomicMax + exp + f32 atomicAdd
//   aggregation          -> wave-per-edge gather + GLOBAL_ATOMIC_ADD_F32
// ---------------------------------------------------------------------------

typedef float v2f __attribute__((ext_vector_type(2)));
typedef float v8f __attribute__((ext_vector_type(8)));

#define IN_CH   128
#define HEADS   4
#define OUT_CH  32
#define NEG_SLOPE 0.2f
#define LDS_STRIDE 132   // 128 + 4 pad: rows land on distinct LDS banks
#define MTILES  4        // 16-row M-tiles per wave -> 64 rows per block

__device__ __forceinline__ float lrelu(float v) {
    return v > 0.0f ? v : NEG_SLOPE * v;
}

// monotone float <-> uint mapping so atomicMax(uint) == float max (any sign)
__device__ __forceinline__ unsigned fkey(float f) {
    unsigned u = __float_as_uint(f);
    return (u & 0x80000000u) ? ~u : (u | 0x80000000u);
}
__device__ __forceinline__ float funkey(unsigned k) {
    return __uint_as_float((k & 0x80000000u) ? (k ^ 0x80000000u) : ~k);
}

// ---------------------------------------------------------------------------
// Kernel 0: init output (bias) and softmax accumulators. Atomics accumulate
// into d_out, so every launch must re-seed it (graph replay safe).
// ---------------------------------------------------------------------------
__global__ void gat_init(float* __restrict__ out, const float* __restrict__ bias,
                         unsigned* __restrict__ mkey, float* __restrict__ denom,
                         int N) {
    int i = blockIdx.x * blockDim.x + threadIdx.x;
    int tot = N * IN_CH;
    if (i < tot) out[i] = bias[i & (IN_CH - 1)];
    if (i < N * HEADS) { mkey[i] = 0u; denom[i] = 0.0f; }
}

// ---------------------------------------------------------------------------
// Kernel 0b: Wt[n][k] = W[k][n]. 64 KB, runs once per launch; makes the two
// B-fragment floats per lane adjacent so the GEMM loads them as one b64.
// ---------------------------------------------------------------------------
__global__ void gat_transpose_W(const float* __restrict__ W, float* __restrict__ Wt) {
    int i = blockIdx.x * blockDim.x + threadIdx.x;   // 0 .. 128*128-1
    int k = i >> 7;
    int n = i & (IN_CH - 1);
    Wt[(size_t)n * IN_CH + k] = W[(size_t)k * IN_CH + n];
}

// ---------------------------------------------------------------------------
// Kernel 1: h = x @ W via V_WMMA_F32_16X16X4_F32.
// Block = 256 threads = 8 waves; block owns 64 rows of x staged in LDS.
// Wave w owns output columns [16w,16w+16) and 4 stacked 16-row M-tiles, so
// each B fragment feeds 4 WMMAs. K=128 stepped by 4.
// A frag (16x4 f32): lanes 0-15 hold K=k0,k0+1; lanes 16-31 hold K=k0+2,k0+3.
// B frag (4x16 f32): mirrored (from Wt, so the 2 values are contiguous).
// C/D: 8 VGPRs, lanes 16-31 are rows M+8.
// ---------------------------------------------------------------------------
__global__ __launch_bounds__(256) void gat_gemm_h(const float* __restrict__ x,
                                                  const float* __restrict__ Wt,
                                                  float* __restrict__ h, int N) {
    __shared__ float As[16 * MTILES * LDS_STRIDE];
    const int tid   = threadIdx.x;
    const int mbase = blockIdx.x * 16 * MTILES;

    // cooperative A-tile load: 64x128 f32 (float4 x 8 per thread)
    for (int i = tid; i < (16 * MTILES * IN_CH) / 4; i += 256) {
        int row = (i * 4) / IN_CH;
        int col = (i * 4) % IN_CH;
        int srow = mbase + row; if (srow >= N) srow = N - 1;   // clamp, keep EXEC full
        float4 v = *(const float4*)(x + (size_t)srow * IN_CH + col);
        float* d = &As[row * LDS_STRIDE + col];
        d[0] = v.x; d[1] = v.y; d[2] = v.z; d[3] = v.w;
    }
    __syncthreads();

    const int wave  = tid >> 5;
    const int lane  = tid & 31;
    const int hi    = lane >> 4;        // half-wave select
    const int lrow  = lane & 15;
    const int nbase = wave * 16;

    v8f acc[MTILES] = {};
    const float* wrow = Wt + (size_t)(nbase + lrow) * IN_CH;   // this lane's B column
    for (int k0 = 0; k0 < IN_CH; k0 += 4) {
        const int ka = k0 + hi * 2;
        v2f b;
        {   // contiguous pair Wt[n][ka], Wt[n][ka+1] -> single b64 load
            float2 bv = *(const float2*)(wrow + ka);
            b.x = bv.x; b.y = bv.y;
        }
#pragma unroll
        for (int t = 0; t < MTILES; ++t) {
            v2f a;
            a.x = As[(t * 16 + lrow) * LDS_STRIDE + ka];
            a.y = As[(t * 16 + lrow) * LDS_STRIDE + ka + 1];
            acc[t] = __builtin_amdgcn_wmma_f32_16x16x4_f32(
                /*neg_a=*/false, a, /*neg_b=*/false, b,
                /*c_mod=*/(short)0, acc[t], /*reuse_a=*/false, /*reuse_b=*/false);
        }
    }

#pragma unroll
    for (int t = 0; t < MTILES; ++t) {
#pragma unroll
        for (int r = 0; r < 8; ++r) {
            int mrow = mbase + t * 16 + r + hi * 8;
            if (mrow < N) h[(size_t)mrow * IN_CH + nbase + lrow] = acc[t][r];
        }
    }
}

// ---------------------------------------------------------------------------
// Kernel 2: a_src[n,h] = sum_c h[n,h,c]*att_src[h,c] (and a_dst).
// One wave per (node, head); 32 lanes = 32 channels; shuffle reduction.
// ---------------------------------------------------------------------------
__global__ __launch_bounds__(256) void gat_attn_sums(const float* __restrict__ h,
                                                     const float* __restrict__ att_src,
                                                     const float* __restrict__ att_dst,
                                                     float* __restrict__ a_src,
                                                     float* __restrict__ a_dst, int N) {
    int gw   = blockIdx.x * 8 + (threadIdx.x >> 5);
    int lane = threadIdx.x & 31;
    int n    = gw >> 2;          // HEADS == 4
    int head = gw & 3;
    if (n >= N) return;
    float hv = h[(size_t)n * IN_CH + head * OUT_CH + lane];
    float s  = hv * att_src[head * OUT_CH + lane];
    float d  = hv * att_dst[head * OUT_CH + lane];
#pragma unroll
    for (int off = 16; off > 0; off >>= 1) {
        s += __shfl_xor(s, off, 32);
        d += __shfl_xor(d, off, 32);
    }
    if (lane == 0) {
        a_src[(size_t)n * HEADS + head] = s;
        a_dst[(size_t)n * HEADS + head] = d;
    }
}

// ---------------------------------------------------------------------------
// Kernel 3: segment max of leaky_relu(a_src[src]+a_dst[dst]) over dst.
// Edges [0,E) from edge_index, [E,E+N) are self loops.
// ---------------------------------------------------------------------------
__global__ void gat_edge_max(const long long* __restrict__ ei, int E, int N,
                             const float* __restrict__ a_src,
                             const float* __restrict__ a_dst,
                             unsigned* __restrict__ mkey) {
    int eid = blockIdx.x * blockDim.x + threadIdx.x;
    int tot = E + N;
    if (eid >= tot) return;
    int s, d;
    if (eid < E) { s = (int)ei[eid]; d = (int)ei[E + eid]; }
    else         { s = d = eid - E; }
    float4 as = *(const float4*)(a_src + (size_t)s * HEADS);
    float4 ad = *(const float4*)(a_dst + (size_t)d * HEADS);
    unsigned* mp = mkey + (size_t)d * HEADS;
    atomicMax(mp + 0, fkey(lrelu(as.x + ad.x)));
    atomicMax(mp + 1, fkey(lrelu(as.y + ad.y)));
    atomicMax(mp + 2, fkey(lrelu(as.z + ad.z)));
    atomicMax(mp + 3, fkey(lrelu(as.w + ad.w)));
}

// ---------------------------------------------------------------------------
// Kernel 4: e = exp(lrelu(..) - max); store numerator, accumulate denominator.
// ---------------------------------------------------------------------------
__global__ void gat_edge_exp(const long long* __restrict__ ei, int E, int N,
                             const float* __restrict__ a_src,
                             const float* __restrict__ a_dst,
                             const unsigned* __restrict__ mkey,
                             float* __restrict__ denom,
                             float* __restrict__ ealpha) {
    int eid = blockIdx.x * blockDim.x + threadIdx.x;
    int tot = E + N;
    if (eid >= tot) return;
    int s, d;
    if (eid < E) { s = (int)ei[eid]; d = (int)ei[E + eid]; }
    else         { s = d = eid - E; }
    float4 as = *(const float4*)(a_src + (size_t)s * HEADS);
    float4 ad = *(const float4*)(a_dst + (size_t)d * HEADS);
    const unsigned* mp = mkey + (size_t)d * HEADS;
    float4 ex;
    ex.x = __expf(lrelu(as.x + ad.x) - funkey(mp[0]));
    ex.y = __expf(lrelu(as.y + ad.y) - funkey(mp[1]));
    ex.z = __expf(lrelu(as.z + ad.z) - funkey(mp[2]));
    ex.w = __expf(lrelu(as.w + ad.w) - funkey(mp[3]));
    *(float4*)(ealpha + (size_t)eid * HEADS) = ex;
    float* dp = denom + (size_t)d * HEADS;
    unsafeAtomicAdd(dp + 0, ex.x);
    unsafeAtomicAdd(dp + 1, ex.y);
    unsafeAtomicAdd(dp + 2, ex.z);
    unsafeAtomicAdd(dp + 3, ex.w);
}

// ---------------------------------------------------------------------------
// Kernel 5: out[dst] += h[src] * alpha. One wave per edge; 4 head-chunks of
// 32 channels. alpha uniform per (edge, head); f32 atomics hit L2 (out fits
// easily in 192 MB L2).
// ---------------------------------------------------------------------------
__global__ __launch_bounds__(256) void gat_scatter(const long long* __restrict__ ei,
                                                   int E, int N,
                                                   const float* __restrict__ hfeat,
                                                   const float* __restrict__ ealpha,
                                                   const float* __restrict__ denom,
                                                   float* __restrict__ out) {
    int eid  = blockIdx.x * 8 + (threadIdx.x >> 5);
    int lane = threadIdx.x & 31;
    int tot  = E + N;
    if (eid >= tot) return;
    int s, d;
    if (eid < E) { s = (int)ei[eid]; d = (int)ei[E + eid]; }
    else         { s = d = eid - E; }
#pragma unroll
    for (int head = 0; head < HEADS; ++head) {
        float alpha = ealpha[(size_t)eid * HEADS + head] /
                      (denom[(size_t)d * HEADS + head] + 1e-16f);
        float v = hfeat[(size_t)s * IN_CH + head * OUT_CH + lane] * alpha;
        unsafeAtomicAdd(out + (size_t)d * IN_CH + head * OUT_CH + lane, v);
    }
}

// ---------------------------------------------------------------------------
extern "C" void kernel_launch(void* const* d_in, const int* in_sizes, int n_in,
                              void* d_out, int out_size, void* d_ws, size_t ws_size,
                              hipStream_t stream) {
    const float*     x        = (const float*)d_in[0];
    const float*     W        = (const float*)d_in[1];
    const float*     att_src  = (const float*)d_in[2];
    const float*     att_dst  = (const float*)d_in[3];
    const float*     bias     = (const float*)d_in[4];
    const long long* ei       = (const long long*)d_in[5];   // int64 per reference
    float*           out      = (float*)d_out;

    const int N = in_sizes[0] / IN_CH;
    const int E = in_sizes[5] / 2;
    const int TOT = E + N;

    // workspace carve-up (256B aligned slices)
    char* ws = (char*)d_ws;
    size_t off = 0;
    auto carve = [&](size_t bytes) {
        char* p = ws + off;
        off = (off + bytes + 255) & ~(size_t)255;
        return p;
    };
    float*    h       = (float*)   carve((size_t)N * IN_CH * sizeof(float));
    float*    Wt      = (float*)   carve((size_t)IN_CH * IN_CH * sizeof(float));
    float*    a_src   = (float*)   carve((size_t)N * HEADS * sizeof(float));
    float*    a_dst   = (float*)   carve((size_t)N * HEADS * sizeof(float));
    unsigned* mkey    = (unsigned*)carve((size_t)N * HEADS * sizeof(unsigned));
    float*    denom   = (float*)   carve((size_t)N * HEADS * sizeof(float));
    float*    ealpha  = (float*)   carve((size_t)TOT * HEADS * sizeof(float));
    (void)ws_size; (void)n_in; (void)out_size;

    gat_init<<<(N * IN_CH + 255) / 256, 256, 0, stream>>>(out, bias, mkey, denom, N);
    gat_transpose_W<<<(IN_CH * IN_CH + 255) / 256, 256, 0, stream>>>(W, Wt);
    gat_gemm_h<<<(N + 16 * MTILES - 1) / (16 * MTILES), 256, 0, stream>>>(x, Wt, h, N);
    gat_attn_sums<<<(N * HEADS + 7) / 8, 256, 0, stream>>>(h, att_src, att_dst,
                                                           a_src, a_dst, N);
    gat_edge_max<<<(TOT + 255) / 256, 256, 0, stream>>>(ei, E, N, a_src, a_dst, mkey);
    gat_edge_exp<<<(TOT + 255) / 256, 256, 0, stream>>>(ei, E, N, a_src, a_dst,
                                                        mkey, denom, ealpha);
    gat_scatter<<<(TOT + 7) / 8, 256, 0, stream>>>(ei, E, N, h, ealpha, denom, out);
}